// MLPAdjacency_5935644803526
// MI455X (gfx1250) — compile-verified
//
#include <hip/hip_runtime.h>
#include <hip/hip_bf16.h>
#include <math.h>

// ---------------------------------------------------------------------------
// Problem: B=4, N=384, DIM=256
//   xp = x @ Wx^T + bx                                   (WMMA f32 GEMM)
//   out[b,i,j] = sigmoid( sum_d relu(xp[b,i,d]+xp[b,j,d]+inc[b,i,j]*Wi[d]+bi[d]) * Ws[d] + bs )
// ---------------------------------------------------------------------------

#define BATCH 4
#define NSEQ  384
#define DIMSZ 256

typedef float v2f __attribute__((ext_vector_type(2)));
typedef float v8f __attribute__((ext_vector_type(8)));

// ---------------------------------------------------------------------------
// Kernel 1: xp[bn][e] = sum_d x[bn][d] * Wx[e][d] + bx[e]
// M = B*N = 1536 rows, N = 256 cols, K = 256.
// One wave32 per 16x16 output tile, V_WMMA_F32_16X16X4_F32 over K in steps of 4.
//
// f32 WMMA fragment layouts (ISA 7.12.2):
//   A 16x4 : lane L (L<16): M=L, K={k,k+1} in v0,v1 ; lanes 16-31: K={k+2,k+3}
//   B 4x16 : lane L (L<16): N=L, K={k,k+1} in v0,v1 ; lanes 16-31: K={k+2,k+3}
//            (B[k][n] = Wx[n][k] -> two consecutive floats of Wx row n)
//   C/D    : VGPR r, lane L -> M = r + 8*(L>=16), N = L%16
// ---------------------------------------------------------------------------
__global__ __launch_bounds__(256) void xproj_wmma_kernel(
    const float* __restrict__ x,   // [1536,256]
    const float* __restrict__ Wx,  // [256,256] row-major [e][d]
    const float* __restrict__ bx,  // [256]
    float* __restrict__ xp)        // [1536,256]
{
    const int lane = threadIdx.x & 31;
    const int wave = threadIdx.x >> 5;
    const int tile = blockIdx.x * 8 + wave;   // 1536 tiles total (96 x 16)
    const int tm   = tile >> 4;               // 0..95  row-tile over 1536
    const int tn   = tile & 15;               // 0..15  col-tile over 256

    const int lmod = lane & 15;
    const int lhi  = lane >> 4;               // 0 or 1

    const int arow = tm * 16 + lmod;          // row of x
    const int bcol = tn * 16 + lmod;          // e (row of Wx)

    const float* aptr = x  + (size_t)arow * DIMSZ + 2 * lhi;
    const float* bptr = Wx + (size_t)bcol * DIMSZ + 2 * lhi;

    v8f acc = {};
#pragma unroll 4
    for (int k = 0; k < DIMSZ; k += 4) {
        v2f a = *(const v2f*)(aptr + k);      // 8B aligned (even element offset)
        v2f b = *(const v2f*)(bptr + k);
        acc = __builtin_amdgcn_wmma_f32_16x16x4_f32(
            /*neg_a=*/false, a, /*neg_b=*/false, b,
            /*c_mod=*/(short)0, acc, /*reuse_a=*/false, /*reuse_b=*/false);
    }

    const float bias = bx[bcol];
    float* op = xp + (size_t)(tm * 16 + lhi * 8) * DIMSZ + bcol;
#pragma unroll
    for (int r = 0; r < 8; ++r)
        op[(size_t)r * DIMSZ] = acc[r] + bias;
}

// ---------------------------------------------------------------------------
// Kernel 2: fused pairwise relu-dot + sigmoid.
// Tile: 16 i-rows x 32 j-cols per 256-thread block (2 outputs / thread).
// LDS:
//   u_s  : xp[i0+r][d]              row-major   (broadcast reads, fp32x4)
//   vT_s : xp[j0+r][d] + bi[d]      transposed [d][j], stride 33 (odd)
//          -> lanes index j at fixed d: banks (33d + j) % 64 conflict-free
//   wi_s / ws_s : Wi, Ws            (broadcast fp32x4 reads)
// ---------------------------------------------------------------------------
#define TI 16
#define TJ 32
#define VSTRIDE (TJ + 1)   // 33

__global__ __launch_bounds__(256) void pairwise_fused_kernel(
    const float* __restrict__ xp,   // [B*N,256]
    const float* __restrict__ inc,  // [B,N,N]
    const float* __restrict__ Wi,   // [256]
    const float* __restrict__ bi,   // [256]
    const float* __restrict__ Ws,   // [256]
    const float* __restrict__ bs,   // [1]
    float* __restrict__ out)        // [B,N,N]
{
    __shared__ __align__(16) float u_s[TI * DIMSZ];        // 16 KB
    __shared__ float vT_s[DIMSZ * VSTRIDE];                // 33 KB
    __shared__ __align__(16) float wi_s[DIMSZ];            // 1 KB
    __shared__ __align__(16) float ws_s[DIMSZ];            // 1 KB

    const int b   = blockIdx.z;
    const int i0  = blockIdx.y * TI;
    const int j0  = blockIdx.x * TJ;
    const int tid = threadIdx.x;

    const float* xpi = xp + ((size_t)b * NSEQ + i0) * DIMSZ;
    const float* xpj = xp + ((size_t)b * NSEQ + j0) * DIMSZ;

    // stage u rows (same layout, contiguous copy)
#pragma unroll
    for (int idx = tid; idx < TI * DIMSZ; idx += 256)
        u_s[idx] = xpi[idx];

    // stage v rows transposed, fold in bi.
    // consecutive tid -> consecutive d, same row: global coalesced,
    // LDS banks (33d + row) % 64 distinct -> conflict-free writes.
#pragma unroll
    for (int idx = tid; idx < TJ * DIMSZ; idx += 256) {
        const int row = idx >> 8;       // j within tile
        const int d   = idx & 255;
        vT_s[d * VSTRIDE + row] = xpj[idx] + bi[d];
    }
    wi_s[tid] = Wi[tid];   // blockDim.x == 256 == DIMSZ
    ws_s[tid] = Ws[tid];
    __syncthreads();

    const int tx = tid & 31;            // j
    const int ty = tid >> 5;            // 0..7 ; i in {ty, ty+8}
    const float bsv = bs[0];

    const size_t pbase = (size_t)b * NSEQ * NSEQ + (size_t)i0 * NSEQ + (size_t)(j0 + tx);
    const float c0 = inc[pbase + (size_t)ty * NSEQ];
    const float c1 = inc[pbase + (size_t)(ty + 8) * NSEQ];

    float a0 = 0.0f, a1 = 0.0f;
#pragma unroll 2
    for (int d = 0; d < DIMSZ; d += 4) {
        const float4 wi4 = *(const float4*)&wi_s[d];
        const float4 ws4 = *(const float4*)&ws_s[d];
        const float4 u0  = *(const float4*)&u_s[ty * DIMSZ + d];
        const float4 u1  = *(const float4*)&u_s[(ty + 8) * DIMSZ + d];
        const float v0 = vT_s[(d + 0) * VSTRIDE + tx];
        const float v1 = vT_s[(d + 1) * VSTRIDE + tx];
        const float v2 = vT_s[(d + 2) * VSTRIDE + tx];
        const float v3 = vT_s[(d + 3) * VSTRIDE + tx];

        a0 = fmaf(fmaxf(fmaf(c0, wi4.x, u0.x + v0), 0.0f), ws4.x, a0);
        a0 = fmaf(fmaxf(fmaf(c0, wi4.y, u0.y + v1), 0.0f), ws4.y, a0);
        a0 = fmaf(fmaxf(fmaf(c0, wi4.z, u0.z + v2), 0.0f), ws4.z, a0);
        a0 = fmaf(fmaxf(fmaf(c0, wi4.w, u0.w + v3), 0.0f), ws4.w, a0);

        a1 = fmaf(fmaxf(fmaf(c1, wi4.x, u1.x + v0), 0.0f), ws4.x, a1);
        a1 = fmaf(fmaxf(fmaf(c1, wi4.y, u1.y + v1), 0.0f), ws4.y, a1);
        a1 = fmaf(fmaxf(fmaf(c1, wi4.z, u1.z + v2), 0.0f), ws4.z, a1);
        a1 = fmaf(fmaxf(fmaf(c1, wi4.w, u1.w + v3), 0.0f), ws4.w, a1);
    }

    const float l0 = a0 + bsv;
    const float l1 = a1 + bsv;
    out[pbase + (size_t)ty * NSEQ]       = 1.0f / (1.0f + __expf(-l0));
    out[pbase + (size_t)(ty + 8) * NSEQ] = 1.0f / (1.0f + __expf(-l1));
}

// ---------------------------------------------------------------------------
// Launch: inputs in order {x, inc, Wx, bx, Wi, bi, Ws, bs}, all fp32.
// Workspace: xp = 1536*256 floats = 1.5 MB.
// ---------------------------------------------------------------------------
extern "C" void kernel_launch(void* const* d_in, const int* in_sizes, int n_in,
                              void* d_out, int out_size, void* d_ws, size_t ws_size,
                              hipStream_t stream) {
    const float* x   = (const float*)d_in[0];
    const float* inc = (const float*)d_in[1];
    const float* Wx  = (const float*)d_in[2];
    const float* bx  = (const float*)d_in[3];
    const float* Wi  = (const float*)d_in[4];
    const float* bi  = (const float*)d_in[5];
    const float* Ws  = (const float*)d_in[6];
    const float* bs  = (const float*)d_in[7];
    float* out = (float*)d_out;
    float* xp  = (float*)d_ws;   // 1536*256 floats

    // Kernel 1: 96*16 = 1536 tiles, 8 waves/block -> 192 blocks.
    xproj_wmma_kernel<<<192, 256, 0, stream>>>(x, Wx, bx, xp);

    // Kernel 2: (384/32) x (384/16) x 4 = 12 x 24 x 4 blocks.
    dim3 grid2(NSEQ / TJ, NSEQ / TI, BATCH);
    pairwise_fused_kernel<<<grid2, 256, 0, stream>>>(xp, inc, Wi, bi, Ws, bs, out);
}